// sconv_30185030156520
// MI455X (gfx1250) — compile-verified
//
#include <hip/hip_runtime.h>
#include <math.h>

typedef float v2f __attribute__((ext_vector_type(2)));
typedef float v8f __attribute__((ext_vector_type(8)));

#define BATCH 8
#define CIN   64
#define COUT  128
#define HH    128
#define WW    128
#define HW    (HH*WW)
#define PPW   32        // pixels per workgroup
#define NTHREADS 256
#define PAD   2         // even pad: keeps 2-dword fragments 8B-aligned, rows odd-bank-offset

__global__ __launch_bounds__(NTHREADS) void sphere_conv_fused(
    const float* __restrict__ x,    const float* __restrict__ lut,
    const float* __restrict__ w1,   const float* __restrict__ b1,
    const float* __restrict__ w2,   const float* __restrict__ b2,
    const float* __restrict__ gamma,const float* __restrict__ beta,
    const float* __restrict__ mean, const float* __restrict__ var,
    float* __restrict__ out)
{
    __shared__ float w2l[COUT][CIN + PAD];   // A tiles: row stride 66 dwords -> bank-conflict-free b64
    __shared__ float w1l[CIN * 9];
    __shared__ float b1l[CIN];
    __shared__ float scl[COUT];              // folded BN scale
    __shared__ float shl[COUT];              // folded BN shift (includes conv bias b2)
    __shared__ int   moff[9][PPW][4];        // clamped corner offsets within an HxW plane
    __shared__ float mwgt[9][PPW][4];        // bilinear weights (zeroed when OOB)
    __shared__ float ylds[PPW][CIN + PAD];   // y, pixel-major: B frags = contiguous aligned b64

    const int tid = threadIdx.x;
    const int gp0 = blockIdx.x * PPW;        // first global pixel of this workgroup

    // ---------------- Phase 1: stage weights, folded BN, bilinear metadata ----------------
    for (int i = tid; i < COUT * CIN; i += NTHREADS)
        w2l[i >> 6][i & 63] = w2[i];
    for (int i = tid; i < CIN * 9; i += NTHREADS) w1l[i] = w1[i];
    if (tid < CIN) b1l[tid] = b1[tid];
    if (tid < COUT) {
        float sc = gamma[tid] * rsqrtf(var[tid] + 1e-5f);
        scl[tid] = sc;
        shl[tid] = beta[tid] - mean[tid] * sc + b2[tid] * sc;
    }

    for (int t = tid; t < 9 * PPW; t += NTHREADS) {
        int s = t / PPW, p = t % PPW;
        int gp = gp0 + p;
        int bb = gp >> 14, hw = gp & (HW - 1);
        long li = ((long)(s * BATCH + bb) * HW + hw) * 2;
        float gx = lut[li], gy = lut[li + 1];
        float ix = (gx + 1.f) * 0.5f * (float)(WW - 1);   // align_corners=True
        float iy = (gy + 1.f) * 0.5f * (float)(HH - 1);
        float x0f = floorf(ix), y0f = floorf(iy);
        float wx1 = ix - x0f, wy1 = iy - y0f;
        float wx0 = 1.f - wx1, wy0 = 1.f - wy1;
        int x0 = (int)x0f, y0 = (int)y0f;
        int x1 = x0 + 1,  y1 = y0 + 1;
        float vx0 = (x0 >= 0 && x0 < WW) ? 1.f : 0.f;
        float vx1 = (x1 >= 0 && x1 < WW) ? 1.f : 0.f;
        float vy0 = (y0 >= 0 && y0 < HH) ? 1.f : 0.f;
        float vy1 = (y1 >= 0 && y1 < HH) ? 1.f : 0.f;
        int x0c = min(max(x0, 0), WW - 1), x1c = min(max(x1, 0), WW - 1);
        int y0c = min(max(y0, 0), HH - 1), y1c = min(max(y1, 0), HH - 1);
        moff[s][p][0] = y0c * WW + x0c;  moff[s][p][1] = y0c * WW + x1c;
        moff[s][p][2] = y1c * WW + x0c;  moff[s][p][3] = y1c * WW + x1c;
        mwgt[s][p][0] = wx0 * wy0 * vx0 * vy0;  mwgt[s][p][1] = wx1 * wy0 * vx1 * vy0;
        mwgt[s][p][2] = wx0 * wy1 * vx0 * vy1;  mwgt[s][p][3] = wx1 * wy1 * vx1 * vy1;
    }
    __syncthreads();

    // ---------------- Phase 2: 9x bilinear gather + grouped 1x1 conv -> y in LDS ----------------
    {
        int p  = tid & (PPW - 1);          // lanes sweep pixels
        int cb = (tid >> 5) * 8;           // each thread handles 8 channels
        int gp = gp0 + p;
        int bb = gp >> 14;
        const float* xb = x + (long)bb * CIN * HW;
        float acc[8];
        #pragma unroll
        for (int j = 0; j < 8; ++j) acc[j] = 0.f;
        for (int s = 0; s < 9; ++s) {
            int   o0 = moff[s][p][0], o1 = moff[s][p][1], o2 = moff[s][p][2], o3 = moff[s][p][3];
            float f0 = mwgt[s][p][0], f1 = mwgt[s][p][1], f2 = mwgt[s][p][2], f3 = mwgt[s][p][3];
            #pragma unroll
            for (int j = 0; j < 8; ++j) {
                const float* xp = xb + (long)(cb + j) * HW;   // gathers stay L2-resident (x = 33.5MB)
                float v = f0 * xp[o0] + f1 * xp[o1] + f2 * xp[o2] + f3 * xp[o3];
                acc[j] += w1l[(cb + j) * 9 + s] * v;
            }
        }
        #pragma unroll
        for (int j = 0; j < 8; ++j) ylds[p][cb + j] = acc[j] + b1l[cb + j];
    }
    __syncthreads();

    // ---------------- Phase 3: pointwise 64->128 via V_WMMA_F32_16X16X4_F32 + BN + ReLU ----------------
    {
        const int wave = tid >> 5, lane = tid & 31;
        const int r = lane & 15, hi = lane >> 4;   // half-wave select
        const int m_base = wave * 16;              // each wave owns 16 output channels
        const int m = m_base + r;

        // A fragments (w2 tile): lane = M row; {VGPR0,VGPR1} = {K, K+1}, half-wave offsets K by +2.
        // Contiguous 8B-aligned LDS -> single ds_load_b64 per fragment.
        v2f afrag[16];
        #pragma unroll
        for (int kb = 0; kb < 16; ++kb)
            afrag[kb] = *(const v2f*)&w2l[m][kb * 4 + 2 * hi];

        #pragma unroll
        for (int nt = 0; nt < 2; ++nt) {           // two 16-pixel N tiles
            int pb = nt * 16;
            v8f c = {0.f, 0.f, 0.f, 0.f, 0.f, 0.f, 0.f, 0.f};
            #pragma unroll
            for (int kb = 0; kb < 16; ++kb) {      // K = 64 in steps of 4
                v2f bfrag = *(const v2f*)&ylds[pb + r][kb * 4 + 2 * hi];
                c = __builtin_amdgcn_wmma_f32_16x16x4_f32(
                        false, afrag[kb], false, bfrag, (short)0, c, false, false);
            }
            int gp = gp0 + pb + r;
            int bb = gp >> 14, hw = gp & (HW - 1);
            #pragma unroll
            for (int i = 0; i < 8; ++i) {
                int o = m_base + i + 8 * hi;       // C/D layout: VGPR i -> M=i (lo half) / M=i+8 (hi half)
                float zv = c[i] * scl[o] + shl[o]; // LDS broadcast reads, no redundant VMEM
                out[(long)bb * COUT * HW + (long)o * HW + hw] = fmaxf(zv, 0.f);
            }
        }
    }
}

extern "C" void kernel_launch(void* const* d_in, const int* in_sizes, int n_in,
                              void* d_out, int out_size, void* d_ws, size_t ws_size,
                              hipStream_t stream) {
    const float* x     = (const float*)d_in[0];
    const float* lut   = (const float*)d_in[1];
    const float* w1    = (const float*)d_in[2];
    const float* b1    = (const float*)d_in[3];
    const float* w2    = (const float*)d_in[4];
    const float* b2    = (const float*)d_in[5];
    const float* gamma = (const float*)d_in[6];
    const float* beta  = (const float*)d_in[7];
    const float* mean  = (const float*)d_in[8];
    const float* var   = (const float*)d_in[9];
    float* out = (float*)d_out;

    int nblocks = (BATCH * HW) / PPW;   // 131072 pixels / 32 = 4096 workgroups
    sphere_conv_fused<<<nblocks, NTHREADS, 0, stream>>>(
        x, lut, w1, b1, w2, b2, gamma, beta, mean, var, out);
}